// GraphAtt_61959198212617
// MI455X (gfx1250) — compile-verified
//
#include <hip/hip_runtime.h>

#define Bc   4
#define Tc   16
#define HWc  4096
#define Cc   64
#define Nc   (Bc * Tc * HWc)     // 262144
#define Ec   1000000
#define EPSc 1e-5f
#define SLOPE 0.01f

#define EDGE_WAVES 8
#define NODE_WAVES 4
#define EROW (Cc + 4)            // 68: stride ≡ 4 (mod 64 banks) -> conflict-free columns
#define NROW (2 * Cc + 4)        // 132

typedef __attribute__((ext_vector_type(2))) float v2f;
typedef __attribute__((ext_vector_type(8))) float v8f;

__device__ __forceinline__ float lrelu(float x) { return x > 0.f ? x : SLOPE * x; }

// ---------------------------------------------------------------------------
// Edge kernel: one wave handles 16 edges.
//   h = lrelu(pos_dist @ W1 + b1)            (scalar, 2xC tiny)
//   d = groupnorm(h @ W2 + b2)               (WMMA f32 16x16x4, K=64)
//   att = (gf[pre] + d) @ Wa + ba            (WMMA f32 16x16x4, K=64)
//   atomicAdd copy[suc] += att * w ; mask[suc]=1
// ---------------------------------------------------------------------------
__global__ __launch_bounds__(32 * EDGE_WAVES)
void edge_kernel(const float* __restrict__ gf, const float* __restrict__ pos,
                 const int* __restrict__ edge, const float* __restrict__ wgt,
                 const float* __restrict__ W1, const float* __restrict__ b1,
                 const float* __restrict__ W2, const float* __restrict__ b2,
                 const float* __restrict__ gdg, const float* __restrict__ gdb,
                 const float* __restrict__ Wa, const float* __restrict__ ba,
                 float* __restrict__ copyb, int* __restrict__ mask)
{
    __shared__ float tile[EDGE_WAVES][16][EROW];
    __shared__ int   sSuc[EDGE_WAVES][16];
    __shared__ float sWgt[EDGE_WAVES][16];

    const int wave = threadIdx.x >> 5;
    const int lane = threadIdx.x & 31;
    const int half = lane >> 4;      // 0: K pair +0/+1, M 0..7 ; 1: K pair +2/+3, M 8..15
    const int row  = lane & 15;

    const int  eBase = (blockIdx.x * EDGE_WAVES + wave) * 16;
    const bool valid = (eBase + row) < Ec;
    const int  eIdx  = valid ? (eBase + row) : (Ec - 1);

    const int eb = edge[eIdx * 4 + 0];
    const int et = edge[eIdx * 4 + 1];
    const int es = edge[eIdx * 4 + 2];
    const int ed = edge[eIdx * 4 + 3];
    const int fpre = (eb * Tc + et) * HWc + es;
    const int fsuc = (eb * Tc + et + 1) * HWc + ed;

    // position is (1,T,HW,2): batch-independent
    const float pd0 = pos[(((et + 1) * HWc + ed) << 1) + 0] - pos[((et * HWc + es) << 1) + 0];
    const float pd1 = pos[(((et + 1) * HWc + ed) << 1) + 1] - pos[((et * HWc + es) << 1) + 1];

    if (half == 0) {
        sSuc[wave][row] = fsuc;
        sWgt[wave][row] = valid ? wgt[eIdx] : 0.f;
    }

    // h tile: each lane fills 32 features of its edge row
    #pragma unroll 8
    for (int j = 0; j < 32; ++j) {
        const int c = half * 32 + j;
        tile[wave][row][c] = lrelu(pd0 * W1[c] + pd1 * W1[Cc + c] + b1[c]);
    }
    __syncthreads();

    const v8f vzero = {0.f, 0.f, 0.f, 0.f, 0.f, 0.f, 0.f, 0.f};

    // ---- GEMM1: (16x64) @ W2 (64x64): 16 K-tiles x 4 N-tiles, fully unrolled
    v2f afrag[16];
    #pragma unroll
    for (int k = 0; k < 16; ++k) {
        afrag[k].x = tile[wave][row][4 * k + 2 * half + 0];
        afrag[k].y = tile[wave][row][4 * k + 2 * half + 1];
    }
    v8f acc[4];
    #pragma unroll
    for (int n = 0; n < 4; ++n) acc[n] = vzero;
    #pragma unroll
    for (int k = 0; k < 16; ++k) {
        #pragma unroll
        for (int n = 0; n < 4; ++n) {
            const int col = 16 * n + row;
            v2f bfrag;
            bfrag.x = W2[(4 * k + 2 * half + 0) * Cc + col];
            bfrag.y = W2[(4 * k + 2 * half + 1) * Cc + col];
            acc[n] = __builtin_amdgcn_wmma_f32_16x16x4_f32(
                false, afrag[k], false, bfrag, (short)0, acc[n], false, false);
        }
    }
    __syncthreads();   // WAR: all lanes done reading h before overwrite
    // D layout: VGPR v -> M = v + 8*half, N = 16n + row
    #pragma unroll
    for (int n = 0; n < 4; ++n) {
        const int col = 16 * n + row;
        const float bb = b2[col];
        #pragma unroll
        for (int v = 0; v < 8; ++v)
            tile[wave][v + 8 * half][col] = acc[n][v] + bb;
    }
    __syncthreads();

    // ---- groupnorm(dist) + add gathered value, in place -------------------
    float s = 0.f, s2 = 0.f;
    #pragma unroll 8
    for (int j = 0; j < 32; ++j) {
        const float x = tile[wave][row][half * 32 + j];
        s += x; s2 += x * x;
    }
    s  += __shfl_xor(s, 16);
    s2 += __shfl_xor(s2, 16);
    const float mean = s * (1.f / Cc);
    const float inv  = rsqrtf(s2 * (1.f / Cc) - mean * mean + EPSc);
    #pragma unroll 8
    for (int j = 0; j < 32; ++j) {
        const int c = half * 32 + j;
        const float de = (tile[wave][row][c] - mean) * inv * gdg[c] + gdb[c];
        tile[wave][row][c] = de + gf[(size_t)fpre * Cc + c];  // value gather (L2 resident)
    }
    __syncthreads();

    // ---- GEMM2: (value + dist_emb) @ Wa, fully unrolled -------------------
    #pragma unroll
    for (int k = 0; k < 16; ++k) {
        afrag[k].x = tile[wave][row][4 * k + 2 * half + 0];
        afrag[k].y = tile[wave][row][4 * k + 2 * half + 1];
    }
    #pragma unroll
    for (int n = 0; n < 4; ++n) acc[n] = vzero;
    #pragma unroll
    for (int k = 0; k < 16; ++k) {
        #pragma unroll
        for (int n = 0; n < 4; ++n) {
            const int col = 16 * n + row;
            v2f bfrag;
            bfrag.x = Wa[(4 * k + 2 * half + 0) * Cc + col];
            bfrag.y = Wa[(4 * k + 2 * half + 1) * Cc + col];
            acc[n] = __builtin_amdgcn_wmma_f32_16x16x4_f32(
                false, afrag[k], false, bfrag, (short)0, acc[n], false, false);
        }
    }

    // ---- scatter att * weight (no-return f32 atomics to L2) ---------------
    #pragma unroll
    for (int n = 0; n < 4; ++n) {
        const int col = 16 * n + row;
        const float bb = ba[col];
        #pragma unroll
        for (int v = 0; v < 8; ++v) {
            const int r = v + 8 * half;
            if (eBase + r < Ec) {
                const float a = (acc[n][v] + bb) * sWgt[wave][r];
                atomicAdd(&copyb[(size_t)sSuc[wave][r] * Cc + col], a);
            }
        }
    }
    if (half == 0 && valid) mask[fsuc] = 1;
}

// ---------------------------------------------------------------------------
// Node kernel: one wave handles 16 nodes.
//   cn   = groupnorm(copy)
//   f    = lrelu(groupnorm(concat[gf, cn] @ Wf + bf))
//   out  = mask ? f : gf
// ---------------------------------------------------------------------------
__global__ __launch_bounds__(32 * NODE_WAVES)
void node_kernel(const float* __restrict__ gf, const float* __restrict__ copyb,
                 const int* __restrict__ mask,
                 const float* __restrict__ gng, const float* __restrict__ gnb,
                 const float* __restrict__ Wf, const float* __restrict__ bf,
                 const float* __restrict__ gfg, const float* __restrict__ gfb,
                 float* __restrict__ out)
{
    __shared__ float tile[NODE_WAVES][16][NROW];

    const int wave = threadIdx.x >> 5;
    const int lane = threadIdx.x & 31;
    const int half = lane >> 4;
    const int row  = lane & 15;
    const int node = (blockIdx.x * NODE_WAVES + wave) * 16 + row;  // N divides evenly

    // concat tile: [0:64) = gf, [64:128) = raw copy (normalized in place below)
    float s = 0.f, s2 = 0.f;
    #pragma unroll 8
    for (int j = 0; j < 32; ++j) {
        const int c = half * 32 + j;
        tile[wave][row][c] = gf[(size_t)node * Cc + c];
        const float cv = copyb[(size_t)node * Cc + c];
        s += cv; s2 += cv * cv;
        tile[wave][row][Cc + c] = cv;
    }
    s  += __shfl_xor(s, 16);
    s2 += __shfl_xor(s2, 16);
    const float mean = s * (1.f / Cc);
    const float inv  = rsqrtf(s2 * (1.f / Cc) - mean * mean + EPSc);
    #pragma unroll 8
    for (int j = 0; j < 32; ++j) {
        const int c = half * 32 + j;
        tile[wave][row][Cc + c] = (tile[wave][row][Cc + c] - mean) * inv * gng[c] + gnb[c];
    }
    __syncthreads();

    // ---- GEMM: (16x128) @ Wf (128x64), 32 K-tiles x 4 N-tiles, unrolled ---
    const v8f vzero = {0.f, 0.f, 0.f, 0.f, 0.f, 0.f, 0.f, 0.f};
    v8f acc[4];
    #pragma unroll
    for (int n = 0; n < 4; ++n) acc[n] = vzero;
    #pragma unroll
    for (int k = 0; k < 32; ++k) {
        v2f afrag;
        afrag.x = tile[wave][row][4 * k + 2 * half + 0];
        afrag.y = tile[wave][row][4 * k + 2 * half + 1];
        #pragma unroll
        for (int n = 0; n < 4; ++n) {
            const int col = 16 * n + row;
            v2f bfrag;
            bfrag.x = Wf[(4 * k + 2 * half + 0) * Cc + col];
            bfrag.y = Wf[(4 * k + 2 * half + 1) * Cc + col];
            acc[n] = __builtin_amdgcn_wmma_f32_16x16x4_f32(
                false, afrag, false, bfrag, (short)0, acc[n], false, false);
        }
    }
    __syncthreads();   // WAR before staging result over the concat tile
    #pragma unroll
    for (int n = 0; n < 4; ++n) {
        const int col = 16 * n + row;
        const float bb = bf[col];
        #pragma unroll
        for (int v = 0; v < 8; ++v)
            tile[wave][v + 8 * half][col] = acc[n][v] + bb;
    }
    __syncthreads();

    // ---- final groupnorm + leaky + mask select ----------------------------
    s = 0.f; s2 = 0.f;
    #pragma unroll 8
    for (int j = 0; j < 32; ++j) {
        const float x = tile[wave][row][half * 32 + j];
        s += x; s2 += x * x;
    }
    s  += __shfl_xor(s, 16);
    s2 += __shfl_xor(s2, 16);
    const float mean2 = s * (1.f / Cc);
    const float inv2  = rsqrtf(s2 * (1.f / Cc) - mean2 * mean2 + EPSc);
    const int m = mask[node];
    #pragma unroll 8
    for (int j = 0; j < 32; ++j) {
        const int c = half * 32 + j;
        const float fx = lrelu((tile[wave][row][c] - mean2) * inv2 * gfg[c] + gfb[c]);
        const float gvc = gf[(size_t)node * Cc + c];      // L1/L2 hit (loaded above)
        __builtin_nontemporal_store(m ? fx : gvc, &out[(size_t)node * Cc + c]);
    }
}

extern "C" void kernel_launch(void* const* d_in, const int* in_sizes, int n_in,
                              void* d_out, int out_size, void* d_ws, size_t ws_size,
                              hipStream_t stream)
{
    (void)in_sizes; (void)n_in; (void)out_size; (void)ws_size;

    const float* gf  = (const float*)d_in[0];
    const float* pos = (const float*)d_in[1];
    const int*   edg = (const int*)  d_in[2];
    const float* wgt = (const float*)d_in[3];
    const float* W1  = (const float*)d_in[4];
    const float* b1  = (const float*)d_in[5];
    const float* W2  = (const float*)d_in[6];
    const float* b2  = (const float*)d_in[7];
    const float* gdg = (const float*)d_in[8];
    const float* gdb = (const float*)d_in[9];
    const float* Wa  = (const float*)d_in[10];
    const float* ba  = (const float*)d_in[11];
    const float* gng = (const float*)d_in[12];
    const float* gnb = (const float*)d_in[13];
    const float* Wf  = (const float*)d_in[14];
    const float* bf  = (const float*)d_in[15];
    const float* gfg = (const float*)d_in[16];
    const float* gfb = (const float*)d_in[17];
    float* out = (float*)d_out;

    float* copyb = (float*)d_ws;
    int*   mask  = (int*)((char*)d_ws + (size_t)Nc * Cc * sizeof(float));

    hipMemsetAsync(copyb, 0, (size_t)Nc * Cc * sizeof(float), stream);
    hipMemsetAsync(mask,  0, (size_t)Nc * sizeof(int), stream);

    const int edgeWavesTotal = (Ec + 15) / 16;                       // 62500
    const int edgeBlocks = (edgeWavesTotal + EDGE_WAVES - 1) / EDGE_WAVES;
    edge_kernel<<<edgeBlocks, 32 * EDGE_WAVES, 0, stream>>>(
        gf, pos, edg, wgt, W1, b1, W2, b2, gdg, gdb, Wa, ba, copyb, mask);

    const int nodeBlocks = Nc / (16 * NODE_WAVES);                   // 4096
    node_kernel<<<nodeBlocks, 32 * NODE_WAVES, 0, stream>>>(
        gf, copyb, mask, gng, gnb, Wf, bf, gfg, gfb, out);
}